// bilateral_context_block_73272142070067
// MI455X (gfx1250) — compile-verified
//
#include <hip/hip_runtime.h>
#include <hip/hip_bf16.h>
#include <cstdint>

// ----------------------------------------------------------------------------
// Bilateral context block, fused fp32 pipeline for MI455X (gfx1250, wave32).
//
// Bandwidth-bound workload (≈4 GFLOP vs ≈GB of naive traffic): strategy is to
// keep f (4MB) / xyz (1.5MB) / idx (8MB) resident in the 192MB L2 and
// RECOMPUTE the scrambled gathers in every BN-stats pass instead of
// materializing [B,C,K,N] intermediates. Training-mode BN forces a
// stats->finalize->apply chain per conv layer.
//
// conv7 (16x32) and conv8 (32x16) are executed on the matrix pipe with
// V_WMMA_F32_16X16X4_F32 (exact fp32), one wave per 16-point tile, weights
// held as A-matrix fragments in registers, activations staged via LDS.
// ----------------------------------------------------------------------------

typedef float v2f __attribute__((ext_vector_type(2)));
typedef float v8f __attribute__((ext_vector_type(8)));

static constexpr int B_  = 2;
static constexpr int N_  = 65536;
static constexpr int K_  = 16;
static constexpr int NK_ = N_ * K_;          // 1048576
static constexpr float EPS_ = 1e-5f;

// workspace layout (float offsets); buffers overlapped in time:
//   z1 (dead after BN1-apply), oe_in (dead after conv7), z8 (written last)
static constexpr size_t OFF_F     = 0;        // B*8*N   = 1048576
static constexpr size_t OFF_Z1    = 1048576;  // B*8*N
static constexpr size_t OFF_OEIN  = 1048576;  // B*32*N  = 4194304
static constexpr size_t OFF_Z8    = 1048576;  // B*32*N
static constexpr size_t OFF_Z7    = 5242880;  // B*16*N  = 2097152
static constexpr size_t OFF_STATS = 7340032;  // 7 layers * 128 floats
// per-layer stats slot: [0..31]=sum  [32..63]=sumsq  [64..95]=scale  [96..127]=shift
// layers: 0=BN1 1=BN2 2=BN3 3=BN4 4=BN5 5=BN7 6=BN8
static constexpr int OUT_SNX = 4194304;       // shifted_neigh_xyz offset in d_out

#define DEV __device__ __forceinline__

// ---------------- reductions ----------------

// full-wave (32 lane) reduce of (s,q)
DEV void wave_reduce2(float& s, float& q) {
#pragma unroll
  for (int m = 16; m >= 1; m >>= 1) {
    s += __shfl_xor(s, m, 32);
    q += __shfl_xor(q, m, 32);
  }
}

// block-level stats commit: per-lane accumulators -> shared -> 1 atomic/ch/block
template <int NCH>
DEV void commit_stats(float* sred, float* st, const float* aS, const float* aQ, int tid) {
  const int lane = tid & 31, wv = tid >> 5;
  __syncthreads();
#pragma unroll
  for (int c = 0; c < NCH; ++c) {
    float s = aS[c], q = aQ[c];
    wave_reduce2(s, q);
    if (lane == 0) {
      sred[wv * 2 * NCH + c]       = s;
      sred[wv * 2 * NCH + NCH + c] = q;
    }
  }
  __syncthreads();
  if (tid < NCH) {
    float s = 0.f, q = 0.f;
#pragma unroll
    for (int w = 0; w < 8; ++w) {
      s += sred[w * 2 * NCH + tid];
      q += sred[w * 2 * NCH + NCH + tid];
    }
    atomicAdd(st + tid, s);
    atomicAdd(st + 32 + tid, q);
  }
}

// ---------------- scrambled gathers (faithful raw reshapes) ----------------
// neigh_feat[b,c,k,n] = f[b, n&7, idx[b, c*131072 + k*8192 + n>>3]]
DEV void gather_nf(const int* __restrict__ idx, const float* __restrict__ f,
                   int b, int k, int n, float* nf) {
  const int m = n & 7;
  const unsigned ibase = (unsigned)k * 8192u + ((unsigned)n >> 3);
  const int* ib   = idx + b * NK_;
  const float* fb = f + (b * 8 + m) * N_;
#pragma unroll
  for (int c = 0; c < 8; ++c) {
    int p = ib[c * 131072u + ibase];
    nf[c] = fb[p];
  }
}
// neigh_xyz[b,c,k,n]: l=c*2^20+k*2^16+n; i=l/3; m=l%3; xyz[b, idx[b,i], m]
DEV void gather_nxyz(const int* __restrict__ idx, const float* __restrict__ xyz,
                     int b, int k, int n, float* nx) {
  const int* ib   = idx + b * NK_;
  const float* xb = xyz + b * N_ * 3;
#pragma unroll
  for (int c = 0; c < 3; ++c) {
    unsigned l = (unsigned)c * 1048576u + (unsigned)k * 65536u + (unsigned)n;
    unsigned i = l / 3u;
    unsigned m = l - i * 3u;
    int p = ib[i];
    nx[c] = xb[(unsigned)p * 3u + m];
  }
}

// ---------------- kernels ----------------

__global__ void k_zero_stats(float* st) {
  int i = threadIdx.x;
  if (i < 7 * 128) st[i] = 0.f;
}

// conv1 (8x32) per point + BN1 raw stats
__global__ __launch_bounds__(256) void k_conv1(const float* __restrict__ feat,
                                               const float* __restrict__ w,
                                               const float* __restrict__ bias,
                                               float* __restrict__ z1,
                                               float* __restrict__ st) {
  __shared__ float sred[8 * 2 * 8];
  const int tid = blockIdx.x * 256 + threadIdx.x;
  const int b = tid >> 16, n = tid & (N_ - 1);
  float x[32];
#pragma unroll
  for (int c = 0; c < 32; ++c) x[c] = feat[(b * 32 + c) * N_ + n];
  float aS[8], aQ[8];
#pragma unroll
  for (int o = 0; o < 8; ++o) {
    float acc = bias[o];
#pragma unroll
    for (int c = 0; c < 32; ++c) acc = fmaf(w[o * 32 + c], x[c], acc);
    z1[(b * 8 + o) * N_ + n] = acc;
    aS[o] = acc;
    aQ[o] = acc * acc;
  }
  commit_stats<8>(sred, st, aS, aQ, threadIdx.x);
}

// finalize BN: scale/shift from raw sums
__global__ void k_fin(float* st, const float* __restrict__ g,
                      const float* __restrict__ be, int nch, float invCnt) {
  int c = threadIdx.x;
  if (c < nch) {
    float mean = st[c] * invCnt;
    float var  = st[32 + c] * invCnt - mean * mean;
    float sc   = g[c] * rsqrtf(var + EPS_);
    st[64 + c] = sc;
    st[96 + c] = be[c] - mean * sc;
  }
}

__global__ __launch_bounds__(256) void k_bn1_apply(const float* __restrict__ z1,
                                                   const float* __restrict__ st,
                                                   float* __restrict__ f) {
  int tid = blockIdx.x * 256 + threadIdx.x;   // B*8*N
  int ch = (tid >> 16) & 7;
  float v = z1[tid] * st[64 + ch] + st[96 + ch];
  f[tid] = fmaxf(v, 0.f);
}

// BN2 raw stats: z2 = conv2(feat_info) over all (b,k,n)   (grid-stride)
__global__ __launch_bounds__(256) void k_bn2_stats(const int* __restrict__ idx,
                                                   const float* __restrict__ f,
                                                   const float* __restrict__ w2,
                                                   const float* __restrict__ b2,
                                                   float* __restrict__ st) {
  __shared__ float sred[8 * 2 * 3];
  float aS[3] = {0.f, 0.f, 0.f}, aQ[3] = {0.f, 0.f, 0.f};
  const int nthr = gridDim.x * 256;
  for (int pos = blockIdx.x * 256 + threadIdx.x; pos < B_ * K_ * N_; pos += nthr) {
    int b = pos >> 20, k = (pos >> 16) & 15, n = pos & (N_ - 1);
    float nf[8], tf[8];
    gather_nf(idx, f, b, k, n, nf);
#pragma unroll
    for (int c = 0; c < 8; ++c) tf[c] = f[(b * 8 + c) * N_ + n];
#pragma unroll
    for (int o = 0; o < 3; ++o) {
      float z = b2[o];
#pragma unroll
      for (int c = 0; c < 8; ++c) z = fmaf(w2[o * 16 + c], nf[c] - tf[c], z);
#pragma unroll
      for (int c = 0; c < 8; ++c) z = fmaf(w2[o * 16 + 8 + c], tf[c], z);
      aS[o] += z;
      aQ[o] += z * z;
    }
  }
  commit_stats<3>(sred, st, aS, aQ, threadIdx.x);
}

// pass4: recompute off2, emit shifted_neigh_xyz to d_out, BN3+BN4 raw stats
__global__ __launch_bounds__(256) void k_pass4(const int* __restrict__ idx,
                                               const float* __restrict__ xyz,
                                               const float* __restrict__ f,
                                               const float* __restrict__ w2,
                                               const float* __restrict__ b2,
                                               const float* __restrict__ st2,
                                               const float* __restrict__ w3,
                                               const float* __restrict__ b3,
                                               const float* __restrict__ w4,
                                               const float* __restrict__ b4,
                                               float* __restrict__ st3,
                                               float* __restrict__ st4,
                                               float* __restrict__ snx_out) {
  __shared__ float sred[8 * 2 * 8];
  float aS3[8] = {}, aQ3[8] = {}, aS4[8] = {}, aQ4[8] = {};
  const int nthr = gridDim.x * 256;
  for (int pos = blockIdx.x * 256 + threadIdx.x; pos < B_ * K_ * N_; pos += nthr) {
    int b = pos >> 20, k = (pos >> 16) & 15, n = pos & (N_ - 1);
    float nf[8], tf[8];
    gather_nf(idx, f, b, k, n, nf);
#pragma unroll
    for (int c = 0; c < 8; ++c) tf[c] = f[(b * 8 + c) * N_ + n];
    float off2[3];
#pragma unroll
    for (int o = 0; o < 3; ++o) {
      float z = b2[o];
#pragma unroll
      for (int c = 0; c < 8; ++c) z = fmaf(w2[o * 16 + c], nf[c] - tf[c], z);
#pragma unroll
      for (int c = 0; c < 8; ++c) z = fmaf(w2[o * 16 + 8 + c], tf[c], z);
      off2[o] = fmaxf(z * st2[64 + o] + st2[96 + o], 0.f);
    }
    float nx[3], tx[3], xi[9];
    gather_nxyz(idx, xyz, b, k, n, nx);
#pragma unroll
    for (int c = 0; c < 3; ++c) tx[c] = xyz[(b * N_ + n) * 3 + c];
#pragma unroll
    for (int c = 0; c < 3; ++c) {
      float s = nx[c] + off2[c];
      snx_out[((b * 3 + c) * 16 + k) * N_ + n] = s;  // output #2
      xi[c] = nx[c] - tx[c];
      xi[3 + c] = s;
      xi[6 + c] = tx[c];
    }
#pragma unroll
    for (int o = 0; o < 8; ++o) {
      float z3 = b3[o], z4 = b4[o];
#pragma unroll
      for (int c = 0; c < 9; ++c) {
        z3 = fmaf(w3[o * 9 + c], xi[c], z3);
        z4 = fmaf(w4[o * 9 + c], xi[c], z4);
      }
      aS3[o] += z3; aQ3[o] += z3 * z3;
      aS4[o] += z4; aQ4[o] += z4 * z4;
    }
  }
  commit_stats<8>(sred, st3, aS3, aQ3, threadIdx.x);
  commit_stats<8>(sred, st4, aS4, aQ4, threadIdx.x);
}

// pass5: BN5 raw stats (z5 = conv5([snf, nf-tf, tf]))
__global__ __launch_bounds__(256) void k_pass5(const int* __restrict__ idx,
                                               const float* __restrict__ xyz,
                                               const float* __restrict__ f,
                                               const float* __restrict__ snx,
                                               const float* __restrict__ w3,
                                               const float* __restrict__ b3,
                                               const float* __restrict__ st3,
                                               const float* __restrict__ w5,
                                               const float* __restrict__ b5,
                                               float* __restrict__ st5) {
  __shared__ float sred[8 * 2 * 8];
  float aS[8] = {}, aQ[8] = {};
  const int nthr = gridDim.x * 256;
  for (int pos = blockIdx.x * 256 + threadIdx.x; pos < B_ * K_ * N_; pos += nthr) {
    int b = pos >> 20, k = (pos >> 16) & 15, n = pos & (N_ - 1);
    float nf[8], tf[8];
    gather_nf(idx, f, b, k, n, nf);
#pragma unroll
    for (int c = 0; c < 8; ++c) tf[c] = f[(b * 8 + c) * N_ + n];
    float nx[3], tx[3], xi[9];
    gather_nxyz(idx, xyz, b, k, n, nx);
#pragma unroll
    for (int c = 0; c < 3; ++c) {
      tx[c] = xyz[(b * N_ + n) * 3 + c];
      xi[c] = nx[c] - tx[c];
      xi[3 + c] = snx[((b * 3 + c) * 16 + k) * N_ + n];
      xi[6 + c] = tx[c];
    }
    float snf[8];
#pragma unroll
    for (int o = 0; o < 8; ++o) {
      float z = b3[o];
#pragma unroll
      for (int c = 0; c < 9; ++c) z = fmaf(w3[o * 9 + c], xi[c], z);
      snf[o] = nf[o] + fmaxf(z * st3[64 + o] + st3[96 + o], 0.f);
    }
#pragma unroll
    for (int o = 0; o < 8; ++o) {
      float z = b5[o];
#pragma unroll
      for (int c = 0; c < 8; ++c) z = fmaf(w5[o * 24 + c], snf[c], z);
#pragma unroll
      for (int c = 0; c < 8; ++c) z = fmaf(w5[o * 24 + 8 + c], nf[c] - tf[c], z);
#pragma unroll
      for (int c = 0; c < 8; ++c) z = fmaf(w5[o * 24 + 16 + c], tf[c], z);
      aS[o] += z;
      aQ[o] += z * z;
    }
  }
  commit_stats<8>(sred, st5, aS, aQ, threadIdx.x);
}

// pass6: per point, loop k: overall_info + conv6 + online softmax over K;
// emits oe_in = [overall_max(16), overall_ws(16)]
__global__ __launch_bounds__(256) void k_pass6(const int* __restrict__ idx,
                                               const float* __restrict__ xyz,
                                               const float* __restrict__ f,
                                               const float* __restrict__ snx,
                                               const float* __restrict__ w3,
                                               const float* __restrict__ b3,
                                               const float* __restrict__ st3,
                                               const float* __restrict__ w4,
                                               const float* __restrict__ b4,
                                               const float* __restrict__ st4,
                                               const float* __restrict__ w5,
                                               const float* __restrict__ b5,
                                               const float* __restrict__ st5,
                                               const float* __restrict__ w6,
                                               const float* __restrict__ b6,
                                               float* __restrict__ oe) {
  const int tid = blockIdx.x * 256 + threadIdx.x;  // B*N threads
  const int b = tid >> 16, n = tid & (N_ - 1);
  float tf[8], tx[3];
#pragma unroll
  for (int c = 0; c < 8; ++c) tf[c] = f[(b * 8 + c) * N_ + n];
#pragma unroll
  for (int c = 0; c < 3; ++c) tx[c] = xyz[(b * N_ + n) * 3 + c];

  float M[16], S[16], Wc[16], mx[16];
#pragma unroll
  for (int o = 0; o < 16; ++o) { M[o] = -INFINITY; S[o] = 0.f; Wc[o] = 0.f; mx[o] = -INFINITY; }

  for (int k = 0; k < K_; ++k) {
    float nf[8], nx[3], xi[9];
    gather_nf(idx, f, b, k, n, nf);
    gather_nxyz(idx, xyz, b, k, n, nx);
#pragma unroll
    for (int c = 0; c < 3; ++c) {
      xi[c] = nx[c] - tx[c];
      xi[3 + c] = snx[((b * 3 + c) * 16 + k) * N_ + n];
      xi[6 + c] = tx[c];
    }
    float oi[16];
#pragma unroll
    for (int o = 0; o < 8; ++o) {  // xyz_encoding = BN4(conv4(xi)) relu
      float z = b4[o];
#pragma unroll
      for (int c = 0; c < 9; ++c) z = fmaf(w4[o * 9 + c], xi[c], z);
      oi[o] = fmaxf(z * st4[64 + o] + st4[96 + o], 0.f);
    }
    float snf[8];
#pragma unroll
    for (int o = 0; o < 8; ++o) {  // off3 -> snf
      float z = b3[o];
#pragma unroll
      for (int c = 0; c < 9; ++c) z = fmaf(w3[o * 9 + c], xi[c], z);
      snf[o] = nf[o] + fmaxf(z * st3[64 + o] + st3[96 + o], 0.f);
    }
#pragma unroll
    for (int o = 0; o < 8; ++o) {  // feat_encoding = BN5(conv5) relu
      float z = b5[o];
#pragma unroll
      for (int c = 0; c < 8; ++c) z = fmaf(w5[o * 24 + c], snf[c], z);
#pragma unroll
      for (int c = 0; c < 8; ++c) z = fmaf(w5[o * 24 + 8 + c], nf[c] - tf[c], z);
#pragma unroll
      for (int c = 0; c < 8; ++c) z = fmaf(w5[o * 24 + 16 + c], tf[c], z);
      oi[8 + o] = fmaxf(z * st5[64 + o] + st5[96 + o], 0.f);
    }
#pragma unroll
    for (int o = 0; o < 16; ++o) {  // k_weights logits + online softmax update
      float kw = b6[o];
#pragma unroll
      for (int c = 0; c < 16; ++c) kw = fmaf(w6[o * 16 + c], oi[c], kw);
      float x = oi[o];
      if (kw > M[o]) {
        float sc = __expf(M[o] - kw);
        S[o] = S[o] * sc + 1.f;
        Wc[o] = Wc[o] * sc + x;
        M[o] = kw;
      } else {
        float e = __expf(kw - M[o]);
        S[o] += e;
        Wc[o] += e * x;
      }
      mx[o] = fmaxf(mx[o], x);
    }
  }
#pragma unroll
  for (int o = 0; o < 16; ++o) {
    oe[(b * 32 + o) * N_ + n]        = mx[o];         // overall_max
    oe[(b * 32 + 16 + o) * N_ + n]   = Wc[o] / S[o];  // overall_ws
  }
}

// ---------------- WMMA fp32 conv (CO x CI) over [B, CI, N] -> [B, CO, N] ----
// One wave per 16-point tile. A = weights (16x4 fp32 fragments, registers),
// B = activations via LDS, C seeded with conv bias. Accumulates BN raw stats.
// Launch MUST be 64 blocks x 128 threads (256 waves, 8192 tiles, uniform 32
// tiles/wave so intra-block __syncthreads stays converged).
template <int CI, int CO, bool INBN>
__global__ __launch_bounds__(128) void k_conv_wmma(const float* __restrict__ zin,
                                                   const float* __restrict__ w,
                                                   const float* __restrict__ bias,
                                                   const float* __restrict__ stIn,
                                                   float* __restrict__ zout,
                                                   float* __restrict__ stOut) {
  constexpr int NT = CO / 16;   // D tiles (output-channel blocks)
  constexpr int NJ = CI / 4;    // K chunks
  __shared__ float lds[4][CI][17];  // +1 pad: conflict-free column reads
  const int lane = threadIdx.x & 31;
  const int wv   = threadIdx.x >> 5;
  const int hi   = (lane >> 4) & 1;  // lanes 16-31 hold K+2 / M+8 halves
  const int ln   = lane & 15;

  // A fragments: 16x4 fp32 layout — lane<16: M=lane, {K,K+1}; lane>=16: M=lane-16, {K+2,K+3}
  v2f wa[NT][NJ];
#pragma unroll
  for (int t = 0; t < NT; ++t)
#pragma unroll
    for (int j = 0; j < NJ; ++j) {
      int row = t * 16 + ln;
      int cc = j * 4 + hi * 2;
      wa[t][j].x = w[row * CI + cc];
      wa[t][j].y = w[row * CI + cc + 1];
    }
  float bo[NT][8];
#pragma unroll
  for (int t = 0; t < NT; ++t)
#pragma unroll
    for (int r = 0; r < 8; ++r) bo[t][r] = bias[t * 16 + r + hi * 8];

  float aS[NT][8], aQ[NT][8];
#pragma unroll
  for (int t = 0; t < NT; ++t)
#pragma unroll
    for (int r = 0; r < 8; ++r) { aS[t][r] = 0.f; aQ[t][r] = 0.f; }

  const int wavesTotal = gridDim.x * 4;
  const int waveId = blockIdx.x * 4 + wv;
  const int TILES = B_ * (N_ / 16);  // 8192
  for (int tile = waveId; tile < TILES; tile += wavesTotal) {
    int b  = tile >> 12;
    int nb = (tile & 4095) << 4;
    // stage 16-point activation tile (optionally BN+relu of zin)
#pragma unroll
    for (int it = 0; it < CI * 16 / 32; ++it) {
      int e = it * 32 + lane;
      int ch = e >> 4, p = e & 15;
      float v = zin[(b * CI + ch) * N_ + nb + p];
      if (INBN) v = fmaxf(v * stIn[64 + ch] + stIn[96 + ch], 0.f);
      lds[wv][ch][p] = v;
    }
    __syncthreads();
    v8f acc[NT];
#pragma unroll
    for (int t = 0; t < NT; ++t)
#pragma unroll
      for (int r = 0; r < 8; ++r) acc[t][r] = bo[t][r];
#pragma unroll
    for (int j = 0; j < NJ; ++j) {
      // B fragment (4x16 fp32): VGPR0 rows {K,K+2}, VGPR1 rows {K+1,K+3}, N=lane&15
      int ch = j * 4 + hi * 2;
      v2f bm;
      bm.x = lds[wv][ch][ln];
      bm.y = lds[wv][ch + 1][ln];
#pragma unroll
      for (int t = 0; t < NT; ++t) {
        acc[t] = __builtin_amdgcn_wmma_f32_16x16x4_f32(
            false, wa[t][j], false, bm, (short)0, acc[t], false, false);
      }
    }
    // D layout: VGPR r -> M=r (lanes 0-15) / M=r+8 (lanes 16-31), N=lane&15
#pragma unroll
    for (int t = 0; t < NT; ++t)
#pragma unroll
      for (int r = 0; r < 8; ++r) {
        int o = t * 16 + r + hi * 8;
        float v = acc[t][r];
        zout[(b * CO + o) * N_ + nb + ln] = v;
        aS[t][r] += v;
        aQ[t][r] += v * v;
      }
    __syncthreads();
  }
  // stats: reduce over the 16 points within each half-wave, 1 atomic/channel/wave
#pragma unroll
  for (int t = 0; t < NT; ++t)
#pragma unroll
    for (int r = 0; r < 8; ++r) {
      float s = aS[t][r], q = aQ[t][r];
#pragma unroll
      for (int m = 8; m >= 1; m >>= 1) {
        s += __shfl_xor(s, m, 32);
        q += __shfl_xor(q, m, 32);
      }
      if (ln == 0) {
        int o = t * 16 + r + hi * 8;
        atomicAdd(stOut + o, s);
        atomicAdd(stOut + 32 + o, q);
      }
    }
}

// final: output_feat = leaky_relu(BN8(z8)), slope 0.01
__global__ __launch_bounds__(256) void k_bn8_apply(const float* __restrict__ z8,
                                                   const float* __restrict__ st,
                                                   float* __restrict__ out) {
  int tid = blockIdx.x * 256 + threadIdx.x;  // B*32*N
  int ch = (tid >> 16) & 31;
  float v = z8[tid] * st[64 + ch] + st[96 + ch];
  out[tid] = v > 0.f ? v : 0.01f * v;
}

// ---------------- host ----------------

extern "C" void kernel_launch(void* const* d_in, const int* in_sizes, int n_in,
                              void* d_out, int out_size, void* d_ws, size_t ws_size,
                              hipStream_t stream) {
  (void)in_sizes; (void)n_in; (void)out_size; (void)ws_size;
  // setup_inputs order: feature, input_xyz, params (insertion order), input_neigh_idx
  const float* feat = (const float*)d_in[0];
  const float* xyz  = (const float*)d_in[1];
  const float* c1w = (const float*)d_in[2],  *c1b = (const float*)d_in[3];
  const float* c1g = (const float*)d_in[4],  *c1be = (const float*)d_in[5];
  const float* c2w = (const float*)d_in[6],  *c2b = (const float*)d_in[7];
  const float* c2g = (const float*)d_in[8],  *c2be = (const float*)d_in[9];
  const float* c3w = (const float*)d_in[10], *c3b = (const float*)d_in[11];
  const float* c3g = (const float*)d_in[12], *c3be = (const float*)d_in[13];
  const float* c4w = (const float*)d_in[14], *c4b = (const float*)d_in[15];
  const float* c4g = (const float*)d_in[16], *c4be = (const float*)d_in[17];
  const float* c5w = (const float*)d_in[18], *c5b = (const float*)d_in[19];
  const float* c5g = (const float*)d_in[20], *c5be = (const float*)d_in[21];
  const float* c6w = (const float*)d_in[22], *c6b = (const float*)d_in[23];
  const float* c7w = (const float*)d_in[24], *c7b = (const float*)d_in[25];
  const float* c7g = (const float*)d_in[26], *c7be = (const float*)d_in[27];
  const float* c8w = (const float*)d_in[28], *c8b = (const float*)d_in[29];
  const float* c8g = (const float*)d_in[30], *c8be = (const float*)d_in[31];
  const int*   idx = (const int*)d_in[32];

  float* outp = (float*)d_out;
  float* wsF  = (float*)d_ws;
  float* fbuf = wsF + OFF_F;
  float* z1   = wsF + OFF_Z1;
  float* oe   = wsF + OFF_OEIN;
  float* z7   = wsF + OFF_Z7;
  float* z8   = wsF + OFF_Z8;
  float* ST[7];
  for (int i = 0; i < 7; ++i) ST[i] = wsF + OFF_STATS + i * 128;
  float* snx = outp + OUT_SNX;

  const float invBN  = 1.f / (float)(B_ * N_);        // BN1/BN7/BN8
  const float invBKN = 1.f / (float)(B_ * K_ * N_);   // BN2..BN5

  k_zero_stats<<<1, 1024, 0, stream>>>(wsF + OFF_STATS);
  k_conv1<<<512, 256, 0, stream>>>(feat, c1w, c1b, z1, ST[0]);
  k_fin<<<1, 32, 0, stream>>>(ST[0], c1g, c1be, 8, invBN);
  k_bn1_apply<<<4096, 256, 0, stream>>>(z1, ST[0], fbuf);
  k_bn2_stats<<<512, 256, 0, stream>>>(idx, fbuf, c2w, c2b, ST[1]);
  k_fin<<<1, 32, 0, stream>>>(ST[1], c2g, c2be, 3, invBKN);
  k_pass4<<<512, 256, 0, stream>>>(idx, xyz, fbuf, c2w, c2b, ST[1],
                                   c3w, c3b, c4w, c4b, ST[2], ST[3], snx);
  k_fin<<<1, 32, 0, stream>>>(ST[2], c3g, c3be, 8, invBKN);
  k_fin<<<1, 32, 0, stream>>>(ST[3], c4g, c4be, 8, invBKN);
  k_pass5<<<512, 256, 0, stream>>>(idx, xyz, fbuf, snx, c3w, c3b, ST[2],
                                   c5w, c5b, ST[4]);
  k_fin<<<1, 32, 0, stream>>>(ST[4], c5g, c5be, 8, invBKN);
  k_pass6<<<512, 256, 0, stream>>>(idx, xyz, fbuf, snx, c3w, c3b, ST[2],
                                   c4w, c4b, ST[3], c5w, c5b, ST[4],
                                   c6w, c6b, oe);
  // conv7: 16x32 on raw oe_in -> z7, BN7 stats  (WMMA fp32)
  k_conv_wmma<32, 16, false><<<64, 128, 0, stream>>>(oe, c7w, c7b, ST[5], z7, ST[5]);
  k_fin<<<1, 32, 0, stream>>>(ST[5], c7g, c7be, 16, invBN);
  // conv8: 32x16 on BN7+relu(z7) -> z8, BN8 stats  (WMMA fp32)
  k_conv_wmma<16, 32, true><<<64, 128, 0, stream>>>(z7, c8w, c8b, ST[5], z8, ST[6]);
  k_fin<<<1, 32, 0, stream>>>(ST[6], c8g, c8be, 32, invBN);
  k_bn8_apply<<<16384, 256, 0, stream>>>(z8, ST[6], outp);
}